// DressedQuantumNet_49495203119307
// MI455X (gfx1250) — compile-verified
//
#include <hip/hip_runtime.h>

typedef __attribute__((ext_vector_type(2))) float v2f;
typedef __attribute__((ext_vector_type(8))) float v8f;

#define N_QUBITS 4
#define Q_DEPTH  6
#define IN_F     64
// wire w <-> bit value (1 << (3-w)) in a 16-state index (wire 0 = MSB, matching
// the reference's (B,2,2,2,2) axis ordering).

__device__ __forceinline__ void ry_layer(float st[16], const float ang[4]) {
#pragma unroll
  for (int w = 0; w < 4; ++w) {
    const float h  = 0.5f * ang[w];
    const float c  = __cosf(h);
    const float sn = __sinf(h);
    const int   m  = 1 << (3 - w);
#pragma unroll
    for (int i = 0; i < 16; ++i) {
      if ((i & m) == 0) {
        const int   j  = i | m;
        const float t0 = st[i], t1 = st[j];
        st[i] = __builtin_fmaf(-sn, t1, c * t0);
        st[j] = __builtin_fmaf( sn, t0, c * t1);
      }
    }
  }
}

__device__ __forceinline__ void cnot_g(float st[16], const int cm, const int tm) {
#pragma unroll
  for (int i = 0; i < 16; ++i) {
    if ((i & cm) != 0 && (i & tm) == 0) {
      const int j = i | tm;
      const float t = st[i]; st[i] = st[j]; st[j] = t;
    }
  }
}

__global__ __launch_bounds__(256)
void dqnet_fused_kernel(const float* __restrict__ X,     // (B, 64)
                        const float* __restrict__ preW,  // (4, 64)
                        const float* __restrict__ preB,  // (4,)
                        const float* __restrict__ qP,    // (24,)
                        const float* __restrict__ postW, // (1, 4)
                        const float* __restrict__ postB, // (1,)
                        float* __restrict__ out,         // (B, 1)
                        const int batch) {
  // One block = 8 waves = 256 samples; one wave = 32 samples (2 WMMA M-tiles).
  __shared__ float ytile[8][32][4];

  const int tid   = threadIdx.x;
  const int wv    = tid >> 5;
  const int lane  = tid & 31;
  const int half  = lane >> 4;   // lane-half selects K sub-pair / M upper 8
  const int n     = lane & 15;   // column of B/D tile; sample-row within half for A
  const int sbase = blockIdx.x * 256 + wv * 32;

  // ---------------- Phase 1: pre-layer GEMM with V_WMMA_F32_16X16X4_F32 -------
  // D(16x16) = A(16x4) x B(4x16) + C, accumulated over 16 K-steps (K total 64).
  // Bias folded into the accumulator init (column n gets preB[n&3]; columns
  // n>=4 are duplicates of n&3 and are never stored).
  const float bias = preB[n & 3];
  v8f c0, c1;
#pragma unroll
  for (int v = 0; v < 8; ++v) { c0[v] = bias; c1[v] = bias; }

  const int last = batch - 1;
  const int r0 = min(sbase + n,      last);  // tile0: samples sbase..sbase+15
  const int r1 = min(sbase + 16 + n, last);  // tile1: samples sbase+16..sbase+31
  const float* __restrict__ wrow = preW + (size_t)(n & 3) * IN_F;

#pragma unroll 4
  for (int k0 = 0; k0 < IN_F; k0 += 4) {
    const int kk = k0 + 2 * half;  // lanes 0-15: K=k0,k0+1 ; lanes 16-31: K=k0+2,k0+3
    const v2f a0 = *(const v2f*)(X + (size_t)r0 * IN_F + kk);
    const v2f a1 = *(const v2f*)(X + (size_t)r1 * IN_F + kk);
    const v2f b  = *(const v2f*)(wrow + kk);      // B[k][n] = preW[n][k]
    c0 = __builtin_amdgcn_wmma_f32_16x16x4_f32(false, a0, false, b, (short)0, c0, false, false);
    c1 = __builtin_amdgcn_wmma_f32_16x16x4_f32(false, a1, false, b, (short)0, c1, false, false);
  }

  // Scatter the 4 valid columns of both D tiles into LDS.
  // D element (VGPR v, lane l) = y[M = v + 8*half (+16 for tile1)][N = n].
  if (n < 4) {
#pragma unroll
    for (int v = 0; v < 8; ++v) {
      ytile[wv][v + 8 * half][n]      = c0[v];
      ytile[wv][v + 8 * half + 16][n] = c1[v];
    }
  }
  __syncthreads();

  // ---------------- Phase 2: per-sample 4-qubit statevector in registers ------
  float ang[4];
#pragma unroll
  for (int w = 0; w < 4; ++w)
    ang[w] = tanhf(ytile[wv][lane][w]) * 1.5707963267948966f;

  float st[16];
#pragma unroll
  for (int i = 0; i < 16; ++i) st[i] = 0.25f;  // H^4 |0000>

  ry_layer(st, ang);

#pragma unroll
  for (int k = 0; k < Q_DEPTH; ++k) {
    cnot_g(st, 8, 4);  // CNOT(0,1)
    cnot_g(st, 2, 1);  // CNOT(2,3)
    cnot_g(st, 4, 2);  // CNOT(1,2)
    float aw[4];
#pragma unroll
    for (int w = 0; w < 4; ++w) aw[w] = qP[k * 4 + w];
    ry_layer(st, aw);
  }

  float p[16];
#pragma unroll
  for (int i = 0; i < 16; ++i) p[i] = st[i] * st[i];

  float res = postB[0];
#pragma unroll
  for (int w = 0; w < 4; ++w) {
    const int m = 1 << (3 - w);
    float ez = 0.0f;
#pragma unroll
    for (int i = 0; i < 16; ++i) ez += (i & m) ? -p[i] : p[i];
    res = __builtin_fmaf(ez, postW[w], res);
  }

  const int s = sbase + lane;
  if (s < batch) out[s] = res;
}

extern "C" void kernel_launch(void* const* d_in, const int* in_sizes, int n_in,
                              void* d_out, int out_size, void* d_ws, size_t ws_size,
                              hipStream_t stream) {
  const float* X     = (const float*)d_in[0];
  const float* preW  = (const float*)d_in[1];
  const float* preB  = (const float*)d_in[2];
  const float* qP    = (const float*)d_in[3];
  const float* postW = (const float*)d_in[4];
  const float* postB = (const float*)d_in[5];
  float* out = (float*)d_out;

  const int batch  = in_sizes[0] / IN_F;     // 262144
  const int blocks = (batch + 255) / 256;    // 256 samples per block (8 waves)
  dqnet_fused_kernel<<<blocks, 256, 0, stream>>>(X, preW, preB, qP, postW, postB,
                                                 out, batch);
}